// DoOperatorLayer_37864431681737
// MI455X (gfx1250) — compile-verified
//
#include <hip/hip_runtime.h>
#include <math.h>

// Problem constants (match the reference)
#define BB 512
#define VV 50
#define HH 768
#define NII 8
#define H2 (2 * HH)   // 1536
#define MM (BB * NII) // 4096 rows

typedef __attribute__((ext_vector_type(2))) float v2f;
typedef __attribute__((ext_vector_type(8))) float v8f;

__device__ __forceinline__ float gelu_exact(float x) {
    return 0.5f * x * (1.0f + erff(x * 0.70710678118654752f));
}

// ---------------------------------------------------------------------------
// 1) Copy variable_states -> d_out (float4 vectorized; 157 MB round trip is
//    the HBM roofline floor for this op)
// ---------------------------------------------------------------------------
__global__ void copy_states_f4(const float4* __restrict__ src,
                               float4* __restrict__ dst, int n4) {
    int t = blockIdx.x * blockDim.x + threadIdx.x;
    if (t < n4) dst[t] = src[t];
}

// ---------------------------------------------------------------------------
// 2) Gather X[m, 0:H] = vs[b, clip(idx), :], X[m, H:2H] = iv[b, i, :]
// ---------------------------------------------------------------------------
__global__ void gather_x_kernel(const float* __restrict__ vs,
                                const float* __restrict__ iv,
                                const int* __restrict__ idx,
                                float* __restrict__ X, int total) {
    int t = blockIdx.x * blockDim.x + threadIdx.x;
    if (t >= total) return;
    int h = t % H2;
    int m = t / H2;
    int b = m / NII;
    int raw = idx[m];
    int ic = raw < 0 ? 0 : (raw >= VV ? VV - 1 : raw);
    float val;
    if (h < HH) val = vs[((size_t)b * VV + ic) * HH + h];
    else        val = iv[(size_t)m * HH + (h - HH)];
    X[t] = val;
}

// ---------------------------------------------------------------------------
// 3/4/5) f32 WMMA GEMM: C[M,768] = act(A[M,K] @ W[K,768] + bias)
// Block = 128 threads = 4 waves; each wave owns a 32(M) x 64(N) tile:
// 8 accumulators, A fragments reused 4x, B fragments reused 2x per K-step.
// N is a compile-time constant so all B-matrix loads are immediate offsets
// from a single per-wave pointer advanced by 4*N floats per K-step.
// Per ISA 7.12.2 f32 layouts:
//   A 16x4:  lane<16 -> M=lane,  {a.x,a.y}={K,K+1};  lane>=16 -> {K+2,K+3}
//   B  4x16: VGPR j: lanes<16 K=j, lanes>=16 K=j+2; N striped over lane&15
//   C 16x16: VGPR r: lanes<16 (M=r, N=lane), lanes>=16 (M=r+8, N=lane-16)
// ---------------------------------------------------------------------------
template <bool GELU_ACT>
__global__ __launch_bounds__(128) void wmma_gemm_f32(
        const float* __restrict__ A,
        const float* __restrict__ W,
        const float* __restrict__ bias,
        float* __restrict__ C,
        int K) {
    constexpr int N = HH;
    const int lane  = threadIdx.x & 31;
    const int wave  = threadIdx.x >> 5;   // 0..3
    const int half  = lane >> 4;          // 0 or 1
    const int row16 = lane & 15;
    const int m0 = blockIdx.y * 32;
    const int n0 = blockIdx.x * 256 + wave * 64;

    const v8f vzero = {};
    v8f acc[2][4];
#pragma unroll
    for (int i = 0; i < 2; ++i)
#pragma unroll
        for (int t = 0; t < 4; ++t)
            acc[i][t] = vzero;

    // Pointer-increment addressing: constant offsets inside the loop.
    const float* aptr0 = A + (size_t)(m0 + row16) * K + 2 * half;      // 8B aligned
    const float* aptr1 = aptr0 + (size_t)16 * K;
    const float* wptr  = W + (size_t)(2 * half) * N + (n0 + row16);

    for (int k = 0; k < K; k += 4) {
        v2f a0 = *(const v2f*)aptr0;
        v2f a1 = *(const v2f*)aptr1;
        v2f b0, b1, b2, b3;
        b0.x = wptr[0];      b0.y = wptr[N];
        b1.x = wptr[16];     b1.y = wptr[16 + N];
        b2.x = wptr[32];     b2.y = wptr[32 + N];
        b3.x = wptr[48];     b3.y = wptr[48 + N];

        acc[0][0] = __builtin_amdgcn_wmma_f32_16x16x4_f32(false, a0, false, b0, (short)0, acc[0][0], false, false);
        acc[1][0] = __builtin_amdgcn_wmma_f32_16x16x4_f32(false, a1, false, b0, (short)0, acc[1][0], false, false);
        acc[0][1] = __builtin_amdgcn_wmma_f32_16x16x4_f32(false, a0, false, b1, (short)0, acc[0][1], false, false);
        acc[1][1] = __builtin_amdgcn_wmma_f32_16x16x4_f32(false, a1, false, b1, (short)0, acc[1][1], false, false);
        acc[0][2] = __builtin_amdgcn_wmma_f32_16x16x4_f32(false, a0, false, b2, (short)0, acc[0][2], false, false);
        acc[1][2] = __builtin_amdgcn_wmma_f32_16x16x4_f32(false, a1, false, b2, (short)0, acc[1][2], false, false);
        acc[0][3] = __builtin_amdgcn_wmma_f32_16x16x4_f32(false, a0, false, b3, (short)0, acc[0][3], false, false);
        acc[1][3] = __builtin_amdgcn_wmma_f32_16x16x4_f32(false, a1, false, b3, (short)0, acc[1][3], false, false);

        aptr0 += 4;
        aptr1 += 4;
        wptr  += (size_t)4 * N;
    }

    // Epilogue: bias + optional exact GELU, scatter per C-layout
#pragma unroll
    for (int i = 0; i < 2; ++i) {
#pragma unroll
        for (int t = 0; t < 4; ++t) {
            const int n = n0 + t * 16 + row16;
            const float bn = bias[n];
            float* ccol = C + (size_t)(m0 + i * 16 + 8 * half) * N + n;
#pragma unroll
            for (int r = 0; r < 8; ++r) {
                float v = acc[i][t][r] + bn;
                if (GELU_ACT) v = gelu_exact(v);
                ccol[(size_t)r * N] = v;
            }
        }
    }
}

// ---------------------------------------------------------------------------
// 6) gate[m] = sigmoid(dot(G1[m,:], Wg2[:,0]) + bg2)   (one block per row)
// ---------------------------------------------------------------------------
__global__ void gate_kernel(const float* __restrict__ G1,
                            const float* __restrict__ Wg2,
                            const float* __restrict__ bg2,
                            float* __restrict__ gate) {
    __shared__ float red[256];
    int m = blockIdx.x;
    const float* row = G1 + (size_t)m * HH;
    float s = 0.f;
    for (int h = threadIdx.x; h < HH; h += 256) s += row[h] * Wg2[h];
    red[threadIdx.x] = s;
    __syncthreads();
    for (int off = 128; off > 0; off >>= 1) {
        if (threadIdx.x < off) red[threadIdx.x] += red[threadIdx.x + off];
        __syncthreads();
    }
    if (threadIdx.x == 0) {
        float x = red[0] + bg2[0];
        gate[m] = 1.0f / (1.0f + expf(-x));
    }
}

// ---------------------------------------------------------------------------
// 7) Ordered scatter: one thread per (b,h); sequential i-loop gives the
//    reference's last-write-wins semantics for duplicate indices, race-free.
// ---------------------------------------------------------------------------
__global__ void scatter_blend_kernel(const float* __restrict__ X,   // [M, 2H]; cols 0..H-1 = orig
                                     const float* __restrict__ iv,  // [M, H]
                                     const float* __restrict__ gate,// [M]
                                     const int* __restrict__ idx,   // [M]
                                     float* __restrict__ out) {     // [B, V, H]
    int t = blockIdx.x * blockDim.x + threadIdx.x;
    if (t >= BB * HH) return;
    int h = t % HH;
    int b = t / HH;
#pragma unroll
    for (int i = 0; i < NII; ++i) {
        int m = b * NII + i;
        int raw = idx[m];
        if (raw >= 0 && raw < VV) {
            float g = gate[m];
            float o = X[(size_t)m * H2 + h];
            float v = iv[(size_t)m * HH + h];
            out[((size_t)b * VV + raw) * HH + h] = o + g * (v - o);
        }
    }
}

// ---------------------------------------------------------------------------
extern "C" void kernel_launch(void* const* d_in, const int* in_sizes, int n_in,
                              void* d_out, int out_size, void* d_ws, size_t ws_size,
                              hipStream_t stream) {
    const float* vs   = (const float*)d_in[0];   // (B,V,H)
    // d_in[1] = edge_probs: unused by the reference output
    const float* iv   = (const float*)d_in[2];   // (B,NI,H)
    const float* W1   = (const float*)d_in[3];   // (2H,H)
    const float* b1   = (const float*)d_in[4];   // (H)
    const float* W2   = (const float*)d_in[5];   // (H,H)
    const float* b2   = (const float*)d_in[6];   // (H)
    const float* Wg1  = (const float*)d_in[7];   // (H,H)
    const float* bg1  = (const float*)d_in[8];   // (H)
    const float* Wg2  = (const float*)d_in[9];   // (H,1)
    const float* bg2  = (const float*)d_in[10];  // (1)
    const int*   idx  = (const int*)d_in[11];    // (B,NI)
    float* out = (float*)d_out;

    // Workspace layout (floats), total ~50.3 MB
    float* ws   = (float*)d_ws;
    float* X    = ws;                                  // M x 2H
    float* Y1   = X + (size_t)MM * H2;                 // M x H (reused as G1)
    float* ENC  = Y1 + (size_t)MM * HH;                // M x H
    float* gate = ENC + (size_t)MM * HH;               // M

    // 1) out = variable_states
    {
        int n4 = (BB * VV * HH) / 4;
        copy_states_f4<<<(n4 + 255) / 256, 256, 0, stream>>>(
            (const float4*)vs, (float4*)out, n4);
    }
    // 2) gather X
    {
        int total = MM * H2;
        gather_x_kernel<<<(total + 255) / 256, 256, 0, stream>>>(vs, iv, idx, X, total);
    }
    // 3) Y1 = gelu(X @ W1 + b1)    [M=4096, N=768, K=1536]
    {
        dim3 grid(HH / 256, MM / 32);
        wmma_gemm_f32<true><<<grid, 128, 0, stream>>>(X, W1, b1, Y1, H2);
    }
    // 4) ENC = Y1 @ W2 + b2        [K=768]
    {
        dim3 grid(HH / 256, MM / 32);
        wmma_gemm_f32<false><<<grid, 128, 0, stream>>>(Y1, W2, b2, ENC, HH);
    }
    // 5) G1 = gelu(ENC @ Wg1 + bg1) -> reuse Y1 buffer  [K=768]
    {
        dim3 grid(HH / 256, MM / 32);
        wmma_gemm_f32<true><<<grid, 128, 0, stream>>>(ENC, Wg1, bg1, Y1, HH);
    }
    // 6) gate = sigmoid(G1 @ Wg2 + bg2)
    gate_kernel<<<MM, 256, 0, stream>>>(Y1, Wg2, bg2, gate);
    // 7) ordered scatter-blend into out
    {
        int total = BB * HH;
        scatter_blend_kernel<<<(total + 255) / 256, 256, 0, stream>>>(X, iv, gate, idx, out);
    }
}